// clDiceLoss_4947802325210
// MI455X (gfx1250) — compile-verified
//
#include <hip/hip_runtime.h>
#include <hip/hip_bf16.h>

#define H 512
#define W 512
#define TILE_OUT 64
#define RHALO 20
#define TILE (TILE_OUT + 2 * RHALO)   // 104
#define TCELLS (TILE * TILE)          // 10816
#define NT 512                        // threads per block (16 wave32s)
#define NTD 256                       // threads per block, dice kernel
#define ITERS 10
#define TPR (W / TILE_OUT)            // 8 tiles per row

// ---- CDNA5 async global->LDS path (guarded so compile never breaks) ----
#if defined(__HIP_DEVICE_COMPILE__) && \
    __has_builtin(__builtin_amdgcn_global_load_async_to_lds_b32) && \
    __has_builtin(__builtin_amdgcn_s_wait_asynccnt)
#define USE_ASYNC_LDS 1
#if __has_builtin(__builtin_amdgcn_global_load_async_to_lds_b128)
#define ASYNC_B128 1
#else
#define ASYNC_B128 0
#endif
#else
#define USE_ASYNC_LDS 0
#define ASYNC_B128 0
#endif

typedef __attribute__((__vector_size__(4 * sizeof(int)))) int i32x4;

__device__ __forceinline__ float waveRed(float v) {
  // wave32 reduction (warpSize == 32 on gfx1250)
  for (int off = 16; off > 0; off >>= 1) v += __shfl_down(v, off, 32);
  return v;
}

// ---------------- dice sums: sum(p*t), sum(p), sum(t) -------------------
__global__ __launch_bounds__(NTD)
void dice_reduce_kernel(const float4* __restrict__ p, const float4* __restrict__ t,
                        double* __restrict__ dacc, int n4) {
  float sp = 0.f, st = 0.f, spt = 0.f;
  const int stride = gridDim.x * blockDim.x;
  for (int i = blockIdx.x * blockDim.x + threadIdx.x; i < n4; i += stride) {
    // speculative prefetch of the next grid-stride chunk (global_prefetch_b8)
    __builtin_prefetch(p + i + stride, 0, 1);
    __builtin_prefetch(t + i + stride, 0, 1);
    float4 a = p[i], b = t[i];
    sp  += a.x + a.y + a.z + a.w;
    st  += b.x + b.y + b.z + b.w;
    spt += a.x * b.x + a.y * b.y + a.z * b.z + a.w * b.w;
  }
  __shared__ float r[3][NTD / 32];
  sp = waveRed(sp); st = waveRed(st); spt = waveRed(spt);
  const int lane = threadIdx.x & 31, wid = threadIdx.x >> 5;
  if (lane == 0) { r[0][wid] = spt; r[1][wid] = sp; r[2][wid] = st; }
  __syncthreads();
  if (threadIdx.x == 0) {
    float a = 0.f, b = 0.f, c = 0.f;
    for (int w = 0; w < NTD / 32; ++w) { a += r[0][w]; b += r[1][w]; c += r[2][w]; }
    atomicAdd(&dacc[0], (double)a);
    atomicAdd(&dacc[1], (double)b);
    atomicAdd(&dacc[2], (double)c);
  }
}

// ------- fused 10-iteration soft_skeletonize + per-image reductions -----
// Computes S = soft_skeletonize(xin) on a 64x64 output tile (104x104 halo
// tile, fully LDS-resident: 3 x 43.3 KB of the WGP's 320 KB), then
// accumulates sum(S) and sum(S*other) per image.
__global__ __launch_bounds__(NT)
void skel_reduce_kernel(const float* __restrict__ xin, const float* __restrict__ other,
                        float* __restrict__ inter_acc, float* __restrict__ cl_acc) {
  __shared__ float sX[TCELLS];
  __shared__ float sA[TCELLS];
  __shared__ float sB[TCELLS];
  __shared__ float r2[2][NT / 32];

  const int tid = threadIdx.x;
  const int tileId = blockIdx.x;
  const int b = blockIdx.y;
  const int tx = (tileId % TPR) * TILE_OUT;
  const int ty = (tileId / TPR) * TILE_OUT;
  const float* img = xin + (size_t)b * (H * W);
  const float INF = __builtin_inff();

  // ---- halo tile load: out-of-image cells held at +inf (min-identity) ----
  // TILE % 4 == 0 and tile origins are 64-aligned, so every 4-wide group is
  // 16B-aligned in both global and LDS, and is fully in-image or fully out.
#if USE_ASYNC_LDS && ASYNC_B128
  for (int g = tid; g < TCELLS / 4; g += NT) {
    const int i = g / (TILE / 4);
    const int j = (g - i * (TILE / 4)) * 4;
    const int gy = ty + i - RHALO, gx = tx + j - RHALO;
    const bool in = ((unsigned)gy < (unsigned)H) & ((unsigned)gx < (unsigned)W);
    if (in) {
      __builtin_amdgcn_global_load_async_to_lds_b128(
          (i32x4*)(img + gy * W + gx), (i32x4*)&sX[i * TILE + j], 0, 0);
    } else {
      *(float4*)&sX[i * TILE + j] = make_float4(INF, INF, INF, INF);
    }
  }
  __builtin_amdgcn_s_wait_asynccnt(0);
#elif USE_ASYNC_LDS
  for (int idx = tid; idx < TCELLS; idx += NT) {
    const int i = idx / TILE, j = idx - i * TILE;
    const int gy = ty + i - RHALO, gx = tx + j - RHALO;
    const bool in = ((unsigned)gy < (unsigned)H) & ((unsigned)gx < (unsigned)W);
    if (in) {
      __builtin_amdgcn_global_load_async_to_lds_b32(
          (int*)(img + gy * W + gx), (int*)&sX[idx], 0, 0);
    } else {
      sX[idx] = INF;
    }
  }
  __builtin_amdgcn_s_wait_asynccnt(0);
#else
  for (int idx = tid; idx < TCELLS; idx += NT) {
    const int i = idx / TILE, j = idx - i * TILE;
    const int gy = ty + i - RHALO, gx = tx + j - RHALO;
    const bool in = ((unsigned)gy < (unsigned)H) & ((unsigned)gx < (unsigned)W);
    sX[idx] = in ? img[gy * W + gx] : INF;
  }
#endif
  __syncthreads();

  // ---- 10 LDS-resident erosion/top-hat iterations (separable 3x3 pools) ----
  for (int it = 0; it < ITERS; ++it) {
    // A = row-min3(X)
    for (int idx = tid; idx < TCELLS; idx += NT) {
      const int i = idx / TILE, j = idx - i * TILE;
      const int jm = j ? idx - 1 : idx;
      const int jp = (j < TILE - 1) ? idx + 1 : idx;
      sA[idx] = fminf(sX[idx], fminf(sX[jm], sX[jp]));
    }
    __syncthreads();
    // B = col-min3(A) == minpool3(X); force out-of-image cells to -inf
    for (int idx = tid; idx < TCELLS; idx += NT) {
      const int i = idx / TILE, j = idx - i * TILE;
      const int im = i ? idx - TILE : idx;
      const int ip = (i < TILE - 1) ? idx + TILE : idx;
      const float v = fminf(sA[idx], fminf(sA[im], sA[ip]));
      const int gy = ty + i - RHALO, gx = tx + j - RHALO;
      const bool in = ((unsigned)gy < (unsigned)H) & ((unsigned)gx < (unsigned)W);
      sB[idx] = in ? v : -INF;
    }
    __syncthreads();
    // A = row-max3(B)
    for (int idx = tid; idx < TCELLS; idx += NT) {
      const int i = idx / TILE, j = idx - i * TILE;
      const int jm = j ? idx - 1 : idx;
      const int jp = (j < TILE - 1) ? idx + 1 : idx;
      sA[idx] = fmaxf(sB[idx], fmaxf(sB[jm], sB[jp]));
    }
    __syncthreads();
    // X = relu(X - relu(col-max3(A) - B)); out-of-image cells stay +inf
    for (int idx = tid; idx < TCELLS; idx += NT) {
      const int i = idx / TILE, j = idx - i * TILE;
      const int im = i ? idx - TILE : idx;
      const int ip = (i < TILE - 1) ? idx + TILE : idx;
      const float mp = fmaxf(sA[idx], fmaxf(sA[im], sA[ip]));
      const float contour = fmaxf(mp - sB[idx], 0.f);
      const int gy = ty + i - RHALO, gx = tx + j - RHALO;
      const bool in = ((unsigned)gy < (unsigned)H) & ((unsigned)gx < (unsigned)W);
      if (in) sX[idx] = fmaxf(sX[idx] - contour, 0.f);
    }
    __syncthreads();
  }

  // ---- per-image partial sums over the exact central 64x64 region ----
  float s_int = 0.f, s_cl = 0.f;
  const float* oimg = other + (size_t)b * (H * W);
  for (int o = tid; o < TILE_OUT * TILE_OUT; o += NT) {
    const int oi = o >> 6, oj = o & 63;
    const float v = sX[(oi + RHALO) * TILE + (oj + RHALO)];
    s_cl += v;
    s_int += v * oimg[(ty + oi) * W + (tx + oj)];
  }
  s_int = waveRed(s_int);
  s_cl = waveRed(s_cl);
  const int lane = tid & 31, wid = tid >> 5;
  if (lane == 0) { r2[0][wid] = s_int; r2[1][wid] = s_cl; }
  __syncthreads();
  if (tid == 0) {
    float a = 0.f, c = 0.f;
    for (int w = 0; w < NT / 32; ++w) { a += r2[0][w]; c += r2[1][w]; }
    atomicAdd(&inter_acc[b], a);
    atomicAdd(&cl_acc[b], c);
  }
}

// ---------------------------- accumulator init --------------------------
__global__ void init_acc_kernel(double* dacc, float* facc) {
  const int t = threadIdx.x;
  if (t < 3) dacc[t] = 0.0;
  if (t < 256) facc[t] = 0.f;
}

// ------------------------------- finalize -------------------------------
__global__ void finalize_kernel(const double* __restrict__ dacc,
                                const float* __restrict__ facc,
                                float* __restrict__ out, int nimg) {
  __shared__ float sprod[64];
  __shared__ float ssum[64];
  const int t = threadIdx.x;  // 64 threads
  float pr = 0.f, su = 0.f;
  if (t < nimg) {
    const float ifl = (facc[t] + 1.f) / (facc[64 + t] + 1.f);
    const float tfl = (facc[128 + t] + 1.f) / (facc[192 + t] + 1.f);
    pr = ifl * tfl;
    su = ifl + tfl;
  }
  sprod[t] = pr;
  ssum[t] = su;
  __syncthreads();
  if (t == 0) {
    float inter = 0.f, den = 0.f;
    for (int i = 0; i < 64; ++i) { inter += sprod[i]; den += ssum[i]; }
    const float cld = 1.f - 2.f * inter / den;
    const double Ipt = dacc[0], P = dacc[1], T = dacc[2];
    const float dice = 1.f - (float)((2.0 * Ipt + 1e-6) / (P + T + 1e-6));
    out[0] = 0.8f * dice + 0.2f * cld;
  }
}

extern "C" void kernel_launch(void* const* d_in, const int* in_sizes, int n_in,
                              void* d_out, int out_size, void* d_ws, size_t ws_size,
                              hipStream_t stream) {
  const float* pred = (const float*)d_in[0];
  const float* target = (const float*)d_in[1];
  const int n = in_sizes[0];
  const int n4 = n / 4;
  const int nimg = n / (H * W);  // 64

  double* dacc = (double*)d_ws;                       // [0]=sum(p*t) [1]=sum(p) [2]=sum(t)
  float* facc = (float*)((char*)d_ws + 32);           // 4 x 64 per-image sums
  float* inter_p = facc;                              // sum(skel(pred) * target)  per image
  float* cl_p    = facc + 64;                         // sum(skel(pred))           per image
  float* inter_t = facc + 128;                        // sum(skel(target) * pred)  per image
  float* cl_t    = facc + 192;                        // sum(skel(target))         per image

  init_acc_kernel<<<1, 256, 0, stream>>>(dacc, facc);
  dice_reduce_kernel<<<4096, NTD, 0, stream>>>((const float4*)pred, (const float4*)target,
                                               dacc, n4);
  const dim3 gskel(TPR * (H / TILE_OUT), nimg);       // 64 tiles x 64 images
  skel_reduce_kernel<<<gskel, NT, 0, stream>>>(pred, target, inter_p, cl_p);
  skel_reduce_kernel<<<gskel, NT, 0, stream>>>(target, pred, inter_t, cl_t);
  finalize_kernel<<<1, 64, 0, stream>>>(dacc, facc, (float*)d_out, nimg);
}